// RelativeMultiHeadAttention_23287312679390
// MI455X (gfx1250) — compile-verified
//
#include <hip/hip_runtime.h>

typedef float v2f __attribute__((ext_vector_type(2)));
typedef float v8f __attribute__((ext_vector_type(8)));
typedef int   v4i __attribute__((ext_vector_type(4)));

// D = A(16x4 f32) * B(4x16 f32) + C  -- wave32 WMMA, f32 exact
#define WMMA_F32_16x16x4(a, b, c) \
  __builtin_amdgcn_wmma_f32_16x16x4_f32(false, (a), false, (b), (short)0, (c), false, false)

// Async global->LDS copy path (CDNA5): guarded so host pass / older toolchains fall back.
#if defined(__HIP_DEVICE_COMPILE__) && __has_builtin(__builtin_amdgcn_global_load_async_to_lds_b128)
#define USE_ASYNC_LDS 1
typedef __attribute__((address_space(1))) v4i* g_v4i_ptr;   // global src (param type per clang)
typedef __attribute__((address_space(3))) v4i* l_v4i_ptr;   // LDS dst
#else
#define USE_ASYNC_LDS 0
#endif

// ---------------------------------------------------------------------------
// Kernel 1: q = q_x[2048,256] @ Wq[256,256]  (f32 WMMA)
// Block = 16 output rows x 256 cols; 8 waves, each wave owns 2 n-tiles of 16.
// ---------------------------------------------------------------------------
__global__ __launch_bounds__(256) void qproj_kernel(const float* __restrict__ qx,
                                                    const float* __restrict__ Wq,
                                                    float* __restrict__ qout) {
  const int mbase = blockIdx.x * 16;
  const int tid   = threadIdx.x;
  const int lane  = tid & 31;
  const int wv    = tid >> 5;     // 0..7
  const int hi    = lane >> 4;    // 0/1 half of wave
  const int l16   = lane & 15;
  const int row   = mbase + l16;  // A-matrix M index for this lane
  const int n0    = wv * 32;

  v8f c0 = {}; v8f c1 = {};
  for (int kb = 0; kb < 256; kb += 4) {
    const int ka = kb + 2 * hi;          // this half-wave's K pair base
    v2f a, b0, b1;
    a.x = qx[row * 256 + ka];
    a.y = qx[row * 256 + ka + 1];
    const int nA = n0 + l16;
    const int nB = n0 + 16 + l16;
    b0.x = Wq[ka * 256 + nA];
    b0.y = Wq[(ka + 1) * 256 + nA];
    b1.x = Wq[ka * 256 + nB];
    b1.y = Wq[(ka + 1) * 256 + nB];
    c0 = WMMA_F32_16x16x4(a, b0, c0);
    c1 = WMMA_F32_16x16x4(a, b1, c1);
  }
  // D layout: VGPR i -> M = i + 8*hi ; N = lane%16
  for (int i = 0; i < 8; ++i) {
    const int r = mbase + i + 8 * hi;
    qout[r * 256 + n0 + l16]      = c0[i];
    qout[r * 256 + n0 + 16 + l16] = c1[i];
  }
}

// ---------------------------------------------------------------------------
// Kernel 2: fused attention per (b,t).  Reassociated dataflow:
//   A2[64,4]   = Wk-slice^T @ q          (tiny; overlaps the async kv copy)
//   scores     = kv[127,64] @ A2         (WMMA, 8 c-tiles across 8 waves)
//   w          = softmax(scores/8)
//   u[4,64]    = w @ kv                  (VALU)
//   out[h,64]  = u[h] @ Wv[:, h*64:...]  (VALU)
// kv slab (32.5 KB) staged once via GLOBAL_LOAD_ASYNC_TO_LDS_B128,
// padded row stride 72 floats (288 B) for 16-B alignment + bank spread.
// ---------------------------------------------------------------------------
__global__ __launch_bounds__(256) void attn_kernel(const float* __restrict__ kvx,
                                                   const float* __restrict__ Wk,
                                                   const float* __restrict__ Wv,
                                                   const float* __restrict__ qin,
                                                   float* __restrict__ out) {
  constexpr int C = 127;   // context length
  constexpr int K = 64;    // kv_dim
  constexpr int KP = 72;   // padded LDS row stride (floats), 288 B (16-B multiple)
  constexpr int HK = 256;  // h*k

  __shared__ float sQ[256];
  __shared__ __align__(16) float sKV[128 * KP];
  __shared__ float sA2[64][4];
  __shared__ float sS[4][128];
  __shared__ float sU[4][64];
  __shared__ float sRed[4][65];

  const int bt  = blockIdx.x;
  const int tid = threadIdx.x;

  // prefetch the (L2-resident) weight rows this thread will walk
  __builtin_prefetch(Wk + (tid >> 2) * HK + (tid & 3) * K, 0, 3);
  __builtin_prefetch(Wv + (tid >> 6) * K + (tid & 63), 0, 3);

  // --- stage q row and kv slab (zero-pad row 127) ---
  sQ[tid] = qin[bt * 256 + tid];
  const float* kvb = kvx + (size_t)bt * (C * K);
#if USE_ASYNC_LDS
  for (int idx4 = tid; idx4 < (C * K) / 4; idx4 += 256) {
    const int c = idx4 >> 4;            // 16 float4 per kv row
    const int i = (idx4 & 15) * 4;
    __builtin_amdgcn_global_load_async_to_lds_b128(
        (g_v4i_ptr)(kvb + idx4 * 4), (l_v4i_ptr)(&sKV[c * KP + i]), 0, 0);
  }
#else
  for (int idx = tid; idx < C * K; idx += 256)
    sKV[(idx >> 6) * KP + (idx & 63)] = kvb[idx];
#endif
  if (tid < K) sKV[C * KP + tid] = 0.0f;

  // --- A2[i][h] = sum_k Wk[i][h*64+k] * q[h*64+k]  (does not touch sKV) ---
  __syncthreads();   // sQ visible (kv copy still outstanding on ASYNCcnt)
  {
    const int i = tid >> 2, h = tid & 3;
    const float* wr = Wk + i * HK + h * K;
    const float* qr = sQ + h * K;
    float acc = 0.f;
    for (int k = 0; k < K; ++k) acc += wr[k] * qr[k];
    sA2[i][h] = acc;
  }
#if USE_ASYNC_LDS
#if __has_builtin(__builtin_amdgcn_s_wait_asynccnt)
  __builtin_amdgcn_s_wait_asynccnt(0);
#else
  asm volatile("s_wait_asynccnt 0x0" ::: "memory");
#endif
#endif
  __syncthreads();   // sA2 + sKV visible

  // --- scores via WMMA: wave wv computes c-tile [wv*16, wv*16+16) ---
  {
    const int lane = tid & 31;
    const int wv   = tid >> 5;
    const int hi   = lane >> 4;
    const int l16  = lane & 15;
    const int cbase = wv * 16;
    v8f acc = {};
    for (int kb = 0; kb < K; kb += 4) {
      const int ka = kb + 2 * hi;
      v2f a, b;
      a.x = sKV[(cbase + l16) * KP + ka];
      a.y = sKV[(cbase + l16) * KP + ka + 1];
      b.x = (l16 < 4) ? sA2[ka][l16] : 0.f;      // N padded 4 -> 16 with zeros
      b.y = (l16 < 4) ? sA2[ka + 1][l16] : 0.f;
      acc = WMMA_F32_16x16x4(a, b, acc);
    }
    if (l16 < 4) {
      for (int i = 0; i < 8; ++i) {
        const int c = cbase + i + 8 * hi;
        sS[l16][c] = acc[i] * 0.125f;            // 1/sqrt(64)
      }
    }
  }
  __syncthreads();

  // --- softmax over c in [0,127); c==127 is the zero pad row, masked ---
  {
    const int h = tid >> 6, t0 = tid & 63;
    const int c2 = t0 + 64;
    const float s1 = sS[h][t0];
    const float s2 = (c2 < C) ? sS[h][c2] : -1e30f;
    sRed[h][t0] = fmaxf(s1, s2);
    __syncthreads();
    for (int s = 32; s > 0; s >>= 1) {
      if (t0 < s) sRed[h][t0] = fmaxf(sRed[h][t0], sRed[h][t0 + s]);
      __syncthreads();
    }
    const float mx = sRed[h][0];
    __syncthreads();
    const float e1 = __expf(s1 - mx);
    const float e2 = (c2 < C) ? __expf(s2 - mx) : 0.f;
    sRed[h][t0] = e1 + e2;
    __syncthreads();
    for (int s = 32; s > 0; s >>= 1) {
      if (t0 < s) sRed[h][t0] += sRed[h][t0 + s];
      __syncthreads();
    }
    const float inv = 1.0f / sRed[h][0];
    sS[h][t0] = e1 * inv;
    sS[h][c2] = e2 * inv;
  }
  __syncthreads();

  // --- u[h][i] = sum_c w[h][c] * kv[c][i] ---
  {
    const int h = tid >> 6, i = tid & 63;
    float acc = 0.f;
    for (int c = 0; c < C; ++c) acc += sS[h][c] * sKV[c * KP + i];
    sU[h][i] = acc;
  }
  __syncthreads();

  // --- out[h*64+k] = sum_i u[h][i] * Wv[i][h*64+k] ---
  {
    const int h = tid >> 6, k = tid & 63;
    float acc = 0.f;
    const float* wvp = Wv + h * K + k;
    for (int i = 0; i < K; ++i) acc += sU[h][i] * wvp[i * HK];
    out[(size_t)bt * 256 + tid] = acc;
  }
}

// ---------------------------------------------------------------------------
extern "C" void kernel_launch(void* const* d_in, const int* in_sizes, int n_in,
                              void* d_out, int out_size, void* d_ws, size_t ws_size,
                              hipStream_t stream) {
  const float* qx  = (const float*)d_in[0];  // [16,128,256]
  const float* kvx = (const float*)d_in[1];  // [16,128,127,64]
  const float* Wq  = (const float*)d_in[2];  // [256,256]
  const float* Wk  = (const float*)d_in[3];  // [64,256]
  const float* Wv  = (const float*)d_in[4];  // [64,256]
  float* outp = (float*)d_out;               // [16,128,256]
  float* qws  = (float*)d_ws;                // 2048*256 f32 = 2 MB scratch

  qproj_kernel<<<128, 256, 0, stream>>>(qx, Wq, qws);
  attn_kernel<<<2048, 256, 0, stream>>>(kvx, Wk, Wv, qws, outp);
}